// Final_PatchExpand2D_24129126269080
// MI455X (gfx1250) — compile-verified
//
#include <hip/hip_runtime.h>
#include <cstdint>

// Problem geometry (fixed by the reference):
//   x: (B=16, C=384, H=64, W=64) fp32 ; E = 96 ; out: (16, 96, 128, 128) fp32
#define BATCH 16
#define CCH   384
#define ECH   96
#define HH    64
#define WW    64
#define HW    (HH * WW)                 // 4096 elements / 16 KB per (b,c) slice

typedef __attribute__((ext_vector_type(2))) float v2f;
typedef __attribute__((ext_vector_type(4))) float v4f;

// ---------------------------------------------------------------------------
// Pass 1: per-(channel,batch) slice sum & sum-of-squares.
// One block per (c,b) slice (6144 blocks x 256 threads = 8 waves).
// Each wave async-copies its 2 KB sub-tile global->LDS (CDNA5 ASYNC path,
// 512 B per global_load_async_to_lds_b128, no VGPR staging), drains ASYNCcnt,
// then reduces from LDS with b128 ds loads + wave32 shuffle tree.
// ---------------------------------------------------------------------------
__global__ void __launch_bounds__(256)
k_slice_reduce(const float* __restrict__ x,
               float* __restrict__ p1, float* __restrict__ p2)
{
    __shared__ float smem[HW];          // 16 KB staging (one slice)
    __shared__ float red1[8], red2[8];

    const unsigned s = blockIdx.x;      // s = c*16 + b
    const unsigned c = s >> 4;
    const unsigned b = s & 15u;
    const float* sbase = x + ((size_t)b * CCH + c) * HW;

    const unsigned wave = threadIdx.x >> 5;
    const unsigned lane = threadIdx.x & 31u;
    const unsigned base_off = (wave << 11) + (lane << 4);        // byte offset
    const unsigned lds_base = (unsigned)(uintptr_t)&smem[0];     // low 32b of flat = LDS offset
    const uint64_t gbase    = (uint64_t)(uintptr_t)sbase;        // uniform -> SGPR pair

    // 4 async 16B/lane transfers per wave: wave covers [wave*2048, wave*2048+2048)
    #pragma unroll
    for (int k = 0; k < 4; ++k) {
        unsigned o    = base_off + (unsigned)k * 512u;
        unsigned lptr = lds_base + o;
        asm volatile("global_load_async_to_lds_b128 %0, %1, %2"
                     :: "v"(lptr), "v"(o), "s"(gbase) : "memory");
    }
    asm volatile("s_wait_asynccnt 0x0" ::: "memory");

    float s1 = 0.f, s2 = 0.f;
    #pragma unroll
    for (int k = 0; k < 4; ++k) {
        v4f d = *(const v4f*)((const char*)smem + base_off + k * 512);
        s1 += (d.x + d.y) + (d.z + d.w);
        s2 = fmaf(d.x, d.x, s2);
        s2 = fmaf(d.y, d.y, s2);
        s2 = fmaf(d.z, d.z, s2);
        s2 = fmaf(d.w, d.w, s2);
    }

    // wave32 butterfly reduction (warpSize == 32 on gfx1250)
    #pragma unroll
    for (int m = 16; m > 0; m >>= 1) {
        s1 += __shfl_xor(s1, m, 32);
        s2 += __shfl_xor(s2, m, 32);
    }
    if (lane == 0) { red1[wave] = s1; red2[wave] = s2; }
    __syncthreads();
    if (threadIdx.x == 0) {
        float t1 = 0.f, t2 = 0.f;
        #pragma unroll
        for (int w = 0; w < 8; ++w) { t1 += red1[w]; t2 += red2[w]; }
        p1[s] = t1;                 // plain overwrite: deterministic, no atomics
        p2[s] = t2;
    }
}

// ---------------------------------------------------------------------------
// Pass 2 (tiny): fold stats + gamma/beta + the IWT 0.5 factor into per-channel
// scale/shift. One block, 384 threads (12 waves).
//   mu[c]   = S1[c]/N                         (N = 16*64*64 = 65536)
//   var[e]  = (1/(4N)) * sum_q (S2[qE+e] - S1[qE+e]^2/N)
//   scale[c]= 0.5*gamma[c]/sqrt(var+eps) ; shift[c]= 0.5*(beta[c]-mu[c]*g/sqrt)
// ---------------------------------------------------------------------------
__global__ void __launch_bounds__(384)
k_stats(const float* __restrict__ p1, const float* __restrict__ p2,
        const float* __restrict__ gamma, const float* __restrict__ beta,
        float* __restrict__ scale, float* __restrict__ shift)
{
    __shared__ float dvar[CCH];
    __shared__ float sinv[ECH];

    const int   c    = threadIdx.x;
    const float invN = 1.0f / 65536.0f;

    float s1 = 0.f, s2 = 0.f;
    #pragma unroll
    for (int b = 0; b < BATCH; ++b) { s1 += p1[c * 16 + b]; s2 += p2[c * 16 + b]; }
    const float mu = s1 * invN;
    dvar[c] = fmaf(-s1, mu, s2);            // S2 - S1^2/N
    __syncthreads();

    if (c < ECH) {
        float v = (dvar[c] + dvar[c + ECH] + dvar[c + 2 * ECH] + dvar[c + 3 * ECH])
                  * (0.25f * invN);
        sinv[c] = 1.0f / sqrtf(v + 1e-5f);
    }
    __syncthreads();

    const float g = gamma[c] * sinv[c % ECH];
    scale[c] = 0.5f * g;
    shift[c] = 0.5f * fmaf(-mu, g, beta[c]);
}

// ---------------------------------------------------------------------------
// Pass 3: fused normalize + inverse Haar. Thread = (b, e, h, pair-of-w).
// Reads one float2 from each of the 4 channel groups (L2-resident after pass 1;
// 192 MB L2 > 100.7 MB input), writes two float4 rows of the 2x2 output blocks
// with NON-TEMPORAL stores so the 100 MB output stream doesn't evict the input.
// Channel index is uniform per block -> scale/shift loads are scalar.
// ---------------------------------------------------------------------------
__global__ void __launch_bounds__(256)
k_norm_iwt(const float* __restrict__ x,
           const float* __restrict__ scale, const float* __restrict__ shift,
           float* __restrict__ out)
{
    const unsigned tid = blockIdx.x * 256u + threadIdx.x;
    const unsigned w2  = tid & 31u;          // pair of input columns
    const unsigned h   = (tid >> 5) & 63u;
    const unsigned be  = tid >> 11;          // uniform within a block
    const unsigned e   = be % 96u;
    const unsigned b   = be / 96u;

    const size_t qs = (size_t)ECH * HW;      // channel-group stride
    const float* p0 = x + ((size_t)b * CCH + e) * HW + h * WW + (w2 << 1);

    v2f x1 = *(const v2f*)(p0);              // yL
    v2f x2 = *(const v2f*)(p0 + qs);         // yHL
    v2f x3 = *(const v2f*)(p0 + 2 * qs);     // yLH
    v2f x4 = *(const v2f*)(p0 + 3 * qs);     // yHH

    const float sc0 = scale[e],           sh0 = shift[e];
    const float sc1 = scale[e + ECH],     sh1 = shift[e + ECH];
    const float sc2 = scale[e + 2 * ECH], sh2 = shift[e + 2 * ECH];
    const float sc3 = scale[e + 3 * ECH], sh3 = shift[e + 3 * ECH];

    // normalized * 0.5 (folded into scale/shift)
    float a0 = fmaf(x1.x, sc0, sh0), a1 = fmaf(x1.y, sc0, sh0);
    float b0 = fmaf(x2.x, sc1, sh1), b1 = fmaf(x2.y, sc1, sh1);
    float c0 = fmaf(x3.x, sc2, sh2), c1 = fmaf(x3.y, sc2, sh2);
    float d0 = fmaf(x4.x, sc3, sh3), d1 = fmaf(x4.y, sc3, sh3);

    // Haar butterfly: u=x1+x2 v=x1-x2 s=x3+x4 t=x3-x4
    float u0 = a0 + b0, v0 = a0 - b0, s0 = c0 + d0, t0 = c0 - d0;
    float u1 = a1 + b1, v1 = a1 - b1, s1 = c1 + d1, t1 = c1 - d1;

    v4f r0 = { v0 - t0, u0 - s0, v1 - t1, u1 - s1 };   // a00,a01 for w, w+1
    v4f r1 = { v0 + t0, u0 + s0, v1 + t1, u1 + s1 };   // a10,a11 for w, w+1

    float* o = out + ((size_t)b * ECH + e) * (4 * HW) + ((h << 1) * (2 * WW)) + (w2 << 2);
    __builtin_nontemporal_store(r0, (v4f*)o);                // row 2h
    __builtin_nontemporal_store(r1, (v4f*)(o + 2 * WW));     // row 2h+1
}

// ---------------------------------------------------------------------------
extern "C" void kernel_launch(void* const* d_in, const int* in_sizes, int n_in,
                              void* d_out, int out_size, void* d_ws, size_t ws_size,
                              hipStream_t stream)
{
    const float* x     = (const float*)d_in[0];
    const float* gamma = (const float*)d_in[1];
    const float* beta  = (const float*)d_in[2];
    float* out = (float*)d_out;

    float* ws    = (float*)d_ws;        // 52 KB used, fully overwritten each call
    float* p1    = ws;                  // [384*16]
    float* p2    = ws + CCH * BATCH;    // [384*16]
    float* scale = ws + 2 * CCH * BATCH;        // [384]
    float* shift = ws + 2 * CCH * BATCH + CCH;  // [384]

    k_slice_reduce<<<CCH * BATCH, 256, 0, stream>>>(x, p1, p2);
    k_stats<<<1, CCH, 0, stream>>>(p1, p2, gamma, beta, scale, shift);

    const unsigned n3 = BATCH * ECH * HH * (WW / 2);   // 3,145,728 threads
    k_norm_iwt<<<n3 / 256, 256, 0, stream>>>(x, scale, shift, out);
}